// AttentionLayer_1752346657290
// MI455X (gfx1250) — compile-verified
//
#include <hip/hip_runtime.h>
#include <math.h>

// ---------------------------------------------------------------------------
// Additive (Bahdanau) multi-head attention for MI455X / gfx1250.
// B=2, L=384, HID=512, H=8, D=64.
//
// Pipeline:
//   1) gemm768x512x512: Q = query@Wq^T+bq, K = key@Wk^T+bk, V = value@Wv^T+bv
//      (fp32 WMMA 16x16x4)
//   2) gemm_head:  tQ[b,h] = Q_head @ W1^T + b1 ; tK[b,h] = K_head @ W2^T + b2
//   3) energy_softmax: e[q,k] = vb + sum_d vw[d]*tanh(tQ[q,d]+tK[k,d]);
//      mask; row softmax -> attention (written to d_out)   [TRANS-bound]
//   4) av_gemm: x[b,h,q,:] = attn @ V_head   (fp32 WMMA)
//   5) gemm768x512x512: out = x @ Wo^T + bo  (fp32 WMMA)
// ---------------------------------------------------------------------------

#define B_ 2
#define L_ 384
#define HID_ 512
#define H_ 8
#define D_ 64
#define BH_ (B_ * H_)
#define ROWS_ (B_ * L_)          // 768

typedef float v2f __attribute__((ext_vector_type(2)));
typedef float v8f __attribute__((ext_vector_type(8)));

#if defined(__has_builtin) && __has_builtin(__builtin_amdgcn_tanhf)
#define TANHF(x) __builtin_amdgcn_tanhf(x)
#else
#define TANHF(x) tanhf(x)
#endif

// WMMA fp32 16x16x4:  D(16x16,f32) = A(16x4,f32) * B(4x16,f32) + C
// A: lanes 0-15 -> M=lane, VGPR0/1 = K 0/1 ; lanes 16-31 -> K 2/3
// B: lanes 0-15 -> N=lane, VGPR0/1 = K 0/1 ; lanes 16-31 -> K 2/3
// C/D: VGPR i -> M = i (lanes 0-15) / i+8 (lanes 16-31), N = lane&15
__device__ __forceinline__ v8f wmma_f32(v2f a, v2f b, v8f c) {
  return __builtin_amdgcn_wmma_f32_16x16x4_f32(false, a, false, b, (short)0, c,
                                               false, false);
}

// ---------------------------------------------------------------------------
// Generic 768x512 <- 768x512 @ (512x512)^T + bias  (torch Linear semantics).
// One workgroup = 8 waves; wave w computes a 16(M) x 64(N) strip (4 accums).
// grid.x = ROWS_/16 = 48.
// ---------------------------------------------------------------------------
__global__ __launch_bounds__(256) void gemm768x512x512(
    const float* __restrict__ X, const float* __restrict__ W,
    const float* __restrict__ bias, float* __restrict__ Out) {
  const int tid = threadIdx.x;
  const int lane = tid & 31, wid = tid >> 5;
  const int m0 = blockIdx.x * 16;
  const int n0 = wid * 64;
  const int mrow = m0 + (lane & 15);
  const int khalf = (lane >> 4) << 1;   // 0 or 2
  const int ncol = lane & 15;

  v8f acc[4];
#pragma unroll
  for (int j = 0; j < 4; ++j) {
    const float bv = bias[n0 + j * 16 + ncol];
#pragma unroll
    for (int i = 0; i < 8; ++i) acc[j][i] = bv;
  }

  const float* xrow = X + (size_t)mrow * HID_;
  for (int k0 = 0; k0 < HID_; k0 += 4) {
    const v2f a = *(const v2f*)(xrow + k0 + khalf);
#pragma unroll
    for (int j = 0; j < 4; ++j) {
      const float* wp = W + (size_t)(n0 + j * 16 + ncol) * HID_ + k0 + khalf;
      const v2f b = *(const v2f*)wp;
      acc[j] = wmma_f32(a, b, acc[j]);
    }
  }

  const int mbase = m0 + ((lane >> 4) << 3);
#pragma unroll
  for (int j = 0; j < 4; ++j)
#pragma unroll
    for (int i = 0; i < 8; ++i)
      Out[(size_t)(mbase + i) * HID_ + n0 + j * 16 + ncol] = acc[j][i];
}

// ---------------------------------------------------------------------------
// Per-head projection: tX[bh, l, :] = Xfull[b*L+l, h*64 : h*64+64] @ Wh^T + bh
// Wh is [64,64] row-major. grid = (3, 16); 8 waves/block, each wave 16x64.
// ---------------------------------------------------------------------------
__global__ __launch_bounds__(256) void gemm_head(
    const float* __restrict__ Xfull, const float* __restrict__ Wh,
    const float* __restrict__ bh_bias, float* __restrict__ Out) {
  const int lane = threadIdx.x & 31, wid = threadIdx.x >> 5;
  const int bh = blockIdx.y;
  const int b = bh >> 3, h = bh & 7;
  const int m0 = (blockIdx.x * 8 + wid) * 16;     // 0..368
  const int khalf = (lane >> 4) << 1;
  const int ncol = lane & 15;

  v8f acc[4];
#pragma unroll
  for (int j = 0; j < 4; ++j) {
    const float bv = bh_bias[j * 16 + ncol];
#pragma unroll
    for (int i = 0; i < 8; ++i) acc[j][i] = bv;
  }

  const float* xrow =
      Xfull + (size_t)(b * L_ + m0 + (lane & 15)) * HID_ + h * D_;
  for (int k0 = 0; k0 < D_; k0 += 4) {
    const v2f a = *(const v2f*)(xrow + k0 + khalf);
#pragma unroll
    for (int j = 0; j < 4; ++j) {
      const v2f bf = *(const v2f*)(Wh + (size_t)(j * 16 + ncol) * D_ + k0 + khalf);
      acc[j] = wmma_f32(a, bf, acc[j]);
    }
  }

  const int mbase = m0 + ((lane >> 4) << 3);
#pragma unroll
  for (int j = 0; j < 4; ++j)
#pragma unroll
    for (int i = 0; i < 8; ++i)
      Out[(size_t)(bh * L_ + mbase + i) * D_ + j * 16 + ncol] = acc[j][i];
}

// ---------------------------------------------------------------------------
// Energy + softmax. grid = (L/16, BH), block = 384 (one thread per k column).
// Each thread keeps its tK row (64 f32) in registers; tQ tile + vw in LDS.
// attention written directly to its d_out slot, layout [b,h,q,k].
// ---------------------------------------------------------------------------
__global__ __launch_bounds__(384) void energy_softmax(
    const float* __restrict__ tQ, const float* __restrict__ tK,
    const int* __restrict__ mask, const float* __restrict__ vw,
    const float* __restrict__ vb, float* __restrict__ attn) {
  __shared__ float tQs[16 * D_];
  __shared__ float vws[D_];
  __shared__ float red[16];

  const int tid = threadIdx.x;            // == k column, 0..383
  const int lane = tid & 31, wid = tid >> 5;
  const int bh = blockIdx.y;
  const int b = bh >> 3;
  const int q0 = blockIdx.x * 16;

  // tK row for this k, in registers (fully unrolled use below).
  float tkreg[D_] __attribute__((aligned(16)));
  const float* tkp = tK + (size_t)(bh * L_ + tid) * D_;
#pragma unroll
  for (int d = 0; d < D_; d += 4)
    *(float4*)(tkreg + d) = *(const float4*)(tkp + d);

  // tQ tile (16 x 64) and vw into LDS.
  const float* tqp = tQ + (size_t)(bh * L_ + q0) * D_;
  for (int idx = tid; idx < 16 * D_; idx += 384) tQs[idx] = tqp[idx];
  if (tid < D_) vws[tid] = vw[tid];
  __syncthreads();

  const bool valid = (mask[b * L_ + tid] != 0);
  const float vb0 = vb[0];
  float* arow = attn + (size_t)(bh * L_ + q0) * L_;

  for (int q = 0; q < 16; ++q) {
    const float* tq = tQs + q * D_;
    float acc = 0.f;
#pragma unroll
    for (int d = 0; d < D_; ++d)
      acc = fmaf(vws[d], TANHF(tq[d] + tkreg[d]), acc);
    const float e = valid ? (acc + vb0) : -1e10f;

    // --- block max over 384 threads ---
    float m = e;
#pragma unroll
    for (int off = 16; off; off >>= 1) m = fmaxf(m, __shfl_xor(m, off, 32));
    if (lane == 0) red[wid] = m;
    __syncthreads();
    if (wid == 0) {
      float t = (lane < 12) ? red[lane] : -3.4e38f;
#pragma unroll
      for (int off = 8; off; off >>= 1) t = fmaxf(t, __shfl_xor(t, off, 32));
      if (lane == 0) red[0] = t;
    }
    __syncthreads();
    const float mx = red[0];
    __syncthreads();

    // --- block sum of exp ---
    const float p = __expf(e - mx);
    float s = p;
#pragma unroll
    for (int off = 16; off; off >>= 1) s += __shfl_xor(s, off, 32);
    if (lane == 0) red[wid] = s;
    __syncthreads();
    if (wid == 0) {
      float t = (lane < 12) ? red[lane] : 0.f;
#pragma unroll
      for (int off = 8; off; off >>= 1) t += __shfl_xor(t, off, 32);
      if (lane == 0) red[0] = t;
    }
    __syncthreads();
    arow[(size_t)q * L_ + tid] = p * (1.f / red[0]);
    __syncthreads();  // red reused next q
  }
}

// ---------------------------------------------------------------------------
// x[b, q, h*64+d] = sum_k attn[bh,q,k] * V[b*L+k, h*64+d].
// grid = (3, 16); 8 waves/block, each wave a 16(q) x 64(d) strip, K=384.
// ---------------------------------------------------------------------------
__global__ __launch_bounds__(256) void av_gemm(
    const float* __restrict__ attn, const float* __restrict__ V,
    float* __restrict__ Xout) {
  const int lane = threadIdx.x & 31, wid = threadIdx.x >> 5;
  const int bh = blockIdx.y;
  const int b = bh >> 3, h = bh & 7;
  const int m0 = (blockIdx.x * 8 + wid) * 16;
  const int khalf = (lane >> 4) << 1;
  const int ncol = lane & 15;

  v8f acc[4];
#pragma unroll
  for (int j = 0; j < 4; ++j)
#pragma unroll
    for (int i = 0; i < 8; ++i) acc[j][i] = 0.f;

  const float* arow = attn + (size_t)(bh * L_ + m0 + (lane & 15)) * L_;
  const float* vbase = V + (size_t)b * L_ * HID_ + h * D_;
  for (int k0 = 0; k0 < L_; k0 += 4) {
    const v2f a = *(const v2f*)(arow + k0 + khalf);
    const float* vr0 = vbase + (size_t)(k0 + khalf) * HID_;
#pragma unroll
    for (int j = 0; j < 4; ++j) {
      const int n = j * 16 + ncol;
      v2f bf;
      bf.x = vr0[n];
      bf.y = vr0[HID_ + n];
      acc[j] = wmma_f32(a, bf, acc[j]);
    }
  }

  const int mbase = m0 + ((lane >> 4) << 3);
#pragma unroll
  for (int j = 0; j < 4; ++j)
#pragma unroll
    for (int i = 0; i < 8; ++i)
      Xout[(size_t)(b * L_ + mbase + i) * HID_ + h * D_ + j * 16 + ncol] =
          acc[j][i];
}

// ---------------------------------------------------------------------------
extern "C" void kernel_launch(void* const* d_in, const int* in_sizes, int n_in,
                              void* d_out, int out_size, void* d_ws,
                              size_t ws_size, hipStream_t stream) {
  const float* query = (const float*)d_in[0];
  const float* key_ = (const float*)d_in[1];
  const float* value = (const float*)d_in[2];
  const int* mask = (const int*)d_in[3];
  const float* Wq = (const float*)d_in[4];
  const float* bq = (const float*)d_in[5];
  const float* Wk = (const float*)d_in[6];
  const float* bk = (const float*)d_in[7];
  const float* Wv = (const float*)d_in[8];
  const float* bv = (const float*)d_in[9];
  const float* Wo = (const float*)d_in[10];
  const float* bo = (const float*)d_in[11];
  const float* W1 = (const float*)d_in[12];
  const float* b1 = (const float*)d_in[13];
  const float* W2 = (const float*)d_in[14];
  const float* b2 = (const float*)d_in[15];
  const float* vw = (const float*)d_in[16];
  const float* vb = (const float*)d_in[17];

  // workspace layout (floats): Q,K,V,tQ,tK,X  -> 6 * 768*512 = 9.4 MB
  float* ws = (float*)d_ws;
  const size_t MAT = (size_t)ROWS_ * HID_;  // 393216
  float* Q = ws;
  float* Kp = ws + MAT;
  float* Vp = ws + 2 * MAT;
  float* tQ = ws + 3 * MAT;
  float* tK = ws + 4 * MAT;
  float* X = ws + 5 * MAT;

  float* outX = (float*)d_out;             // [B, L, HID]
  float* outAttn = outX + MAT;             // [B, H, L, L]

  gemm768x512x512<<<48, 256, 0, stream>>>(query, Wq, bq, Q);
  gemm768x512x512<<<48, 256, 0, stream>>>(key_, Wk, bk, Kp);
  gemm768x512x512<<<48, 256, 0, stream>>>(value, Wv, bv, Vp);

  gemm_head<<<dim3(3, BH_), 256, 0, stream>>>(Q, W1, b1, tQ);
  gemm_head<<<dim3(3, BH_), 256, 0, stream>>>(Kp, W2, b2, tK);

  energy_softmax<<<dim3(L_ / 16, BH_), 384, 0, stream>>>(tQ, tK, mask, vw, vb,
                                                         outAttn);

  av_gemm<<<dim3(3, BH_), 256, 0, stream>>>(outAttn, Vp, X);

  gemm768x512x512<<<48, 256, 0, stream>>>(X, Wo, bo, outX);
}